// QuickshiftGroups_63840393887937
// MI455X (gfx1250) — compile-verified
//
#include <hip/hip_runtime.h>
#include <math.h>
#include <stdint.h>

#define HH   224
#define WW   224
#define NB   8
#define NC   3
#define HWSZ (HH*WW)
#define MAXSEG 16
#define WRAD 9                 // quickshift neighbor radius (ceil(3*kernel_size))
#define TS   16
#define HALO (TS + 2*WRAD)     // 34
#define NT   (HH/TS)           // 14 tiles per dim
#define BIGV 1.0e6f

typedef float v2f __attribute__((ext_vector_type(2)));
typedef float v8f __attribute__((ext_vector_type(8)));

// --- CDNA5 async global->LDS path (guarded: falls back to plain loads) ------
#if defined(__has_builtin)
#  if __has_builtin(__builtin_amdgcn_global_load_async_to_lds_b32)
#    define USE_ASYNC_LDS 1
#  endif
#endif
#ifndef USE_ASYNC_LDS
#  define USE_ASYNC_LDS 0
#endif

#if USE_ASYNC_LDS
typedef __attribute__((address_space(1))) int gint;   // global (AS1) int
typedef __attribute__((address_space(3))) int lint;   // LDS    (AS3) int

__device__ __forceinline__ gint* as_global(const void* p) {
    return (gint*)(uintptr_t)p;
}
__device__ __forceinline__ lint* as_lds(void* p) {
    // generic LDS pointer: low 32 bits are the wave-relative LDS byte offset
    return (lint*)(unsigned int)(uintptr_t)p;
}
__device__ __forceinline__ void wait_async_zero() {
#if __has_builtin(__builtin_amdgcn_s_wait_asynccnt)
    __builtin_amdgcn_s_wait_asynccnt(0);
#else
    asm volatile("s_wait_asynccnt 0x0" ::: "memory");
#endif
}
#endif

// ---------------------------------------------------------------------------
// 1) Build 224x224 blur band-matrix with symmetric-reflect boundary baked in.
//    K[i,j] = sum_t k[t] * [reflect(i+t)==j], t=-4..4, gaussian sigma=1.
// ---------------------------------------------------------------------------
__global__ void k_build_kmat(float* __restrict__ Km) {
    int idx = blockIdx.x * blockDim.x + threadIdx.x;
    if (idx >= HH * HH) return;
    int i = idx / HH, j = idx % HH;
    float kt[9], ksum = 0.f;
    #pragma unroll
    for (int t = 0; t < 9; ++t) {
        float d = (float)(t - 4);
        kt[t] = expf(-0.5f * d * d);
        ksum += kt[t];
    }
    float acc = 0.f;
    #pragma unroll
    for (int t = 0; t < 9; ++t) {
        int p = i + (t - 4);
        p = (p < 0) ? (-1 - p) : ((p >= HH) ? (2 * HH - 1 - p) : p);
        if (p == j) acc += kt[t] / ksum;
    }
    Km[idx] = acc;
}

// ---------------------------------------------------------------------------
// 2) Blur pass over rows (axis 0):  dst_plane = K @ src_plane, banded WMMA.
//    One wave32 per 16x16 C-tile; band limits k-blocks to I-1..I+1.
// ---------------------------------------------------------------------------
__global__ __launch_bounds__(256) void k_blur_rows(const float* __restrict__ src,
                                                   const float* __restrict__ Km,
                                                   float* __restrict__ dst) {
    int wave = threadIdx.x >> 5;
    int lane = threadIdx.x & 31;
    int gid  = blockIdx.x * (blockDim.x >> 5) + wave;   // tile id in [0, NT*NT*NB*NC)
    int tile = gid % (NT * NT);
    int bc   = gid / (NT * NT);
    int I = tile / NT, J = tile % NT;
    const float* sp = src + (size_t)bc * HWSZ;
    float*       dp = dst + (size_t)bc * HWSZ;

    int m  = lane & 15;        // M (for A) / N (for B,C)
    int kh = lane >> 4;        // K-half selector
    v8f acc = {};

    int kb0 = (I > 0) ? I - 1 : 0;
    int kb1 = (I < NT - 1) ? I + 1 : NT - 1;
    for (int kb = kb0; kb <= kb1; ++kb) {
        #pragma unroll
        for (int s = 0; s < 4; ++s) {
            int k0 = kb * 16 + s * 4 + 2 * kh;
            v2f a, b;
            a.x = Km[(I * 16 + m) * WW + k0];
            a.y = Km[(I * 16 + m) * WW + k0 + 1];
            b.x = sp[(k0    ) * WW + J * 16 + m];
            b.y = sp[(k0 + 1) * WW + J * 16 + m];
            acc = __builtin_amdgcn_wmma_f32_16x16x4_f32(false, a, false, b,
                                                        (short)0, acc, false, false);
        }
    }
    #pragma unroll
    for (int v = 0; v < 8; ++v)
        dp[(I * 16 + v + 8 * kh) * WW + J * 16 + m] = acc[v];
}

// ---------------------------------------------------------------------------
// 3) Blur pass over columns (axis 1): dst = src @ K^T, banded WMMA (band on J).
// ---------------------------------------------------------------------------
__global__ __launch_bounds__(256) void k_blur_cols(const float* __restrict__ src,
                                                   const float* __restrict__ Km,
                                                   float* __restrict__ dst) {
    int wave = threadIdx.x >> 5;
    int lane = threadIdx.x & 31;
    int gid  = blockIdx.x * (blockDim.x >> 5) + wave;
    int tile = gid % (NT * NT);
    int bc   = gid / (NT * NT);
    int I = tile / NT, J = tile % NT;
    const float* sp = src + (size_t)bc * HWSZ;
    float*       dp = dst + (size_t)bc * HWSZ;

    int m  = lane & 15;
    int kh = lane >> 4;
    v8f acc = {};

    int kb0 = (J > 0) ? J - 1 : 0;
    int kb1 = (J < NT - 1) ? J + 1 : NT - 1;
    for (int kb = kb0; kb <= kb1; ++kb) {
        #pragma unroll
        for (int s = 0; s < 4; ++s) {
            int k0 = kb * 16 + s * 4 + 2 * kh;
            v2f a, b;
            a.x = sp[(I * 16 + m) * WW + k0];       // A[m,k] = src[i, k]
            a.y = sp[(I * 16 + m) * WW + k0 + 1];
            b.x = Km[(J * 16 + m) * WW + k0];       // B[k,n] = K^T[k,n] = K[n,k]
            b.y = Km[(J * 16 + m) * WW + k0 + 1];
            acc = __builtin_amdgcn_wmma_f32_16x16x4_f32(false, a, false, b,
                                                        (short)0, acc, false, false);
        }
    }
    #pragma unroll
    for (int v = 0; v < 8; ++v)
        dp[(I * 16 + v + 8 * kh) * WW + J * 16 + m] = acc[v];
}

// ---------------------------------------------------------------------------
// 4) Density: KDE over 19x19 window, LDS-tiled halo (OOB color = BIG -> exp==0).
//    Halo staged with async global->LDS copies (ASYNCcnt) when available.
// ---------------------------------------------------------------------------
__global__ __launch_bounds__(256) void k_density(const float* __restrict__ img,
                                                 float* __restrict__ dens) {
    __shared__ float lc[NC][HALO][HALO];
    int b  = blockIdx.z;
    int r0 = blockIdx.y * TS, c0 = blockIdx.x * TS;
    int tid = threadIdx.y * TS + threadIdx.x;
    for (int idx = tid; idx < HALO * HALO; idx += TS * TS) {
        int rr = idx / HALO, cc = idx % HALO;
        int gr = r0 + rr - WRAD, gc = c0 + cc - WRAD;
        bool in = (gr >= 0) && (gr < HH) && (gc >= 0) && (gc < WW);
#if USE_ASYNC_LDS
        if (in) {
            #pragma unroll
            for (int ch = 0; ch < NC; ++ch)
                __builtin_amdgcn_global_load_async_to_lds_b32(
                    as_global(img + ((size_t)(b * NC + ch) * HH + gr) * WW + gc),
                    as_lds(&lc[ch][rr][cc]), 0, 0);
        } else {
            #pragma unroll
            for (int ch = 0; ch < NC; ++ch)
                lc[ch][rr][cc] = BIGV;
        }
#else
        #pragma unroll
        for (int ch = 0; ch < NC; ++ch)
            lc[ch][rr][cc] = in ? img[((size_t)(b * NC + ch) * HH + gr) * WW + gc] : BIGV;
#endif
    }
#if USE_ASYNC_LDS
    wait_async_zero();
#endif
    __syncthreads();

    int ly = threadIdx.y + WRAD, lx = threadIdx.x + WRAD;
    float p0 = lc[0][ly][lx], p1 = lc[1][ly][lx], p2 = lc[2][ly][lx];
    const float inv = -0.5f / 9.0f;
    float acc = 0.f;
    for (int dr = -WRAD; dr <= WRAD; ++dr) {
        for (int dc = -WRAD; dc <= WRAD; ++dc) {
            float d0 = p0 - lc[0][ly + dr][lx + dc];
            float d1 = p1 - lc[1][ly + dr][lx + dc];
            float d2 = p2 - lc[2][ly + dr][lx + dc];
            float d = d0 * d0 + d1 * d1 + d2 * d2 + (float)(dr * dr + dc * dc);
            acc += __expf(d * inv);
        }
    }
    dens[(size_t)b * HWSZ + (r0 + threadIdx.y) * WW + (c0 + threadIdx.x)] = acc;
}

// ---------------------------------------------------------------------------
// 5) Parent: nearest strictly-denser neighbor (scan order = dr outer, dc inner),
//    then cut links with dist^2 > MAX_DIST^2 (=100).
// ---------------------------------------------------------------------------
__global__ __launch_bounds__(256) void k_parent(const float* __restrict__ img,
                                                const float* __restrict__ dens,
                                                int* __restrict__ par) {
    __shared__ float lc[NC][HALO][HALO];
    __shared__ float ld[HALO][HALO];
    int b  = blockIdx.z;
    int r0 = blockIdx.y * TS, c0 = blockIdx.x * TS;
    int tid = threadIdx.y * TS + threadIdx.x;
    for (int idx = tid; idx < HALO * HALO; idx += TS * TS) {
        int rr = idx / HALO, cc = idx % HALO;
        int gr = r0 + rr - WRAD, gc = c0 + cc - WRAD;
        bool in = (gr >= 0) && (gr < HH) && (gc >= 0) && (gc < WW);
#if USE_ASYNC_LDS
        if (in) {
            #pragma unroll
            for (int ch = 0; ch < NC; ++ch)
                __builtin_amdgcn_global_load_async_to_lds_b32(
                    as_global(img + ((size_t)(b * NC + ch) * HH + gr) * WW + gc),
                    as_lds(&lc[ch][rr][cc]), 0, 0);
            __builtin_amdgcn_global_load_async_to_lds_b32(
                as_global(dens + (size_t)b * HWSZ + (size_t)gr * WW + gc),
                as_lds(&ld[rr][cc]), 0, 0);
        } else {
            #pragma unroll
            for (int ch = 0; ch < NC; ++ch)
                lc[ch][rr][cc] = BIGV;
            ld[rr][cc] = -__builtin_inff();
        }
#else
        #pragma unroll
        for (int ch = 0; ch < NC; ++ch)
            lc[ch][rr][cc] = in ? img[((size_t)(b * NC + ch) * HH + gr) * WW + gc] : BIGV;
        ld[rr][cc] = in ? dens[(size_t)b * HWSZ + gr * WW + gc] : -__builtin_inff();
#endif
    }
#if USE_ASYNC_LDS
    wait_async_zero();
#endif
    __syncthreads();

    int ly = threadIdx.y + WRAD, lx = threadIdx.x + WRAD;
    int r = r0 + threadIdx.y, c = c0 + threadIdx.x;
    float p0 = lc[0][ly][lx], p1 = lc[1][ly][lx], p2 = lc[2][ly][lx];
    float myd = ld[ly][lx];
    float bestd = __builtin_inff();
    int self = r * WW + c;
    int bestp = self;
    for (int dr = -WRAD; dr <= WRAD; ++dr) {
        for (int dc = -WRAD; dc <= WRAD; ++dc) {
            float nd = ld[ly + dr][lx + dc];
            float d0 = p0 - lc[0][ly + dr][lx + dc];
            float d1 = p1 - lc[1][ly + dr][lx + dc];
            float d2 = p2 - lc[2][ly + dr][lx + dc];
            float dist = d0 * d0 + d1 * d1 + d2 * d2 + (float)(dr * dr + dc * dc);
            bool better = (nd > myd) && (dist < bestd);
            bestd = better ? dist : bestd;
            bestp = better ? ((r + dr) * WW + (c + dc)) : bestp;
        }
    }
    if (bestd > 100.0f) bestp = self;   // MAX_DIST^2
    par[(size_t)b * HWSZ + self] = bestp;
}

// ---------------------------------------------------------------------------
// 6) Flatten: chase each pixel's parent chain to its root (pure gather,
//    deterministic; chain is acyclic since parent density strictly higher).
// ---------------------------------------------------------------------------
__global__ void k_flatten(const int* __restrict__ par, int* __restrict__ root) {
    int gid = blockIdx.x * blockDim.x + threadIdx.x;
    if (gid >= NB * HWSZ) return;
    int b = gid / HWSZ, i = gid % HWSZ;
    const int* p = par + (size_t)b * HWSZ;
    int cur = p[i];
    for (int it = 0; it < HWSZ; ++it) {
        int nxt = p[cur];
        if (nxt == cur) break;
        cur = nxt;
    }
    root[gid] = cur;
}

// ---------------------------------------------------------------------------
// 7) Per-image inclusive scan of is_root flags (rank roots in index order).
//    One 1024-thread block per image; 49 chunks of 1024 (49*1024 == 50176).
// ---------------------------------------------------------------------------
__global__ __launch_bounds__(1024) void k_scan(const int* __restrict__ root,
                                               int* __restrict__ cum,
                                               int* __restrict__ nseg) {
    __shared__ int sm[1024];
    int b = blockIdx.x, tid = threadIdx.x;
    int carry = 0;
    for (int ch = 0; ch < HWSZ / 1024; ++ch) {
        int idx = ch * 1024 + tid;
        int flag = (root[(size_t)b * HWSZ + idx] == idx) ? 1 : 0;
        sm[tid] = flag;
        __syncthreads();
        for (int off = 1; off < 1024; off <<= 1) {
            int v = (tid >= off) ? sm[tid - off] : 0;
            __syncthreads();
            sm[tid] += v;
            __syncthreads();
        }
        cum[(size_t)b * HWSZ + idx] = sm[tid] + carry;
        carry += sm[1023];
        __syncthreads();
    }
    if (tid == 0) nseg[b] = carry;
}

// ---------------------------------------------------------------------------
// 8) Label (rank-1), clamp to 16 segments via floor(label/(n/16)), one-hot.
// ---------------------------------------------------------------------------
__global__ void k_final(const int* __restrict__ root, const int* __restrict__ cum,
                        const int* __restrict__ nseg, float* __restrict__ out) {
    int gid = blockIdx.x * blockDim.x + threadIdx.x;
    if (gid >= NB * HWSZ) return;
    int b = gid / HWSZ, i = gid % HWSZ;
    int lab = cum[(size_t)b * HWSZ + root[gid]] - 1;
    int n = nseg[b];
    if (n >= MAXSEG) {
        float div = (float)n / (float)MAXSEG;
        lab = (int)floorf((float)lab / div);
    }
    float* ob = out + (size_t)b * MAXSEG * HWSZ;
    #pragma unroll
    for (int s = 0; s < MAXSEG; ++s)
        ob[(size_t)s * HWSZ + i] = (s == lab) ? 1.0f : 0.0f;
}

// ---------------------------------------------------------------------------
extern "C" void kernel_launch(void* const* d_in, const int* in_sizes, int n_in,
                              void* d_out, int out_size, void* d_ws, size_t ws_size,
                              hipStream_t stream) {
    (void)in_sizes; (void)n_in; (void)out_size; (void)ws_size;
    const float* x = (const float*)d_in[0];        // (8,3,224,224) f32
    float* out = (float*)d_out;                    // (8,16,224,224) f32

    float* Km   = (float*)d_ws;                              // 224*224
    float* tmp  = Km   + HWSZ;                               // 8*3*224*224
    float* blur = tmp  + (size_t)NB * NC * HWSZ;             // 8*3*224*224
    float* dens = blur + (size_t)NB * NC * HWSZ;             // 8*224*224
    int*   par  = (int*)(dens + (size_t)NB * HWSZ);          // 8*224*224
    int*   root = par  + (size_t)NB * HWSZ;
    int*   cum  = root + (size_t)NB * HWSZ;
    int*   nseg = cum  + (size_t)NB * HWSZ;                  // 8

    k_build_kmat<<<(HH * HH + 255) / 256, 256, 0, stream>>>(Km);

    int tiles = NT * NT * NB * NC;                 // 14*14*24 = 4704 (divisible by 8)
    k_blur_rows<<<tiles / 8, 256, 0, stream>>>(x, Km, tmp);
    k_blur_cols<<<tiles / 8, 256, 0, stream>>>(tmp, Km, blur);

    dim3 tb(TS, TS);
    dim3 tg(NT, NT, NB);
    k_density<<<tg, tb, 0, stream>>>(blur, dens);
    k_parent<<<tg, tb, 0, stream>>>(blur, dens, par);

    int n = NB * HWSZ;
    k_flatten<<<(n + 255) / 256, 256, 0, stream>>>(par, root);
    k_scan<<<NB, 1024, 0, stream>>>(root, cum, nseg);
    k_final<<<(n + 255) / 256, 256, 0, stream>>>(root, cum, nseg, out);
}